// SynthesisLayer_79817672229329
// MI455X (gfx1250) — compile-verified
//
#include <hip/hip_runtime.h>

// Fused StyleGAN3 filtered_lrelu for MI455X (gfx1250).
// up2(FIR12) -> bias+lrelu*sqrt2+clamp -> down2(FIR12), one WG per (n,c) slice.
// All FIR stages are expressed as banded 16x16 matrix multiplies executed with
// chained V_WMMA_F32_16X16X4_F32 (fp32 matrix pipe, full precision).

typedef float v2f __attribute__((ext_vector_type(2)));
typedef float v8f __attribute__((ext_vector_type(8)));

#define P1 65    // pitch (floats) for X and T1 (width 64)
#define P2 145   // pitch (floats) for T2 and T3 (width 144)
#define WAVES 8

// LDS layout (floats):
//   [0,     9280)  X (64 x P1, only 64x65=4160 used) -> reused as T3 (64 x P2)
//   [9280, 18640)  T1 (144 x P1)
//   [18640,39520)  T2 (144 x P2)
//   [39520,39544)  filters: f[12] (up), g[12] (down)
#define OFF_T1 9280
#define OFF_T2 18640
#define OFF_FG 39520
#define SMEM_FLOATS 39552   // 158,208 bytes < 160 KB

static __device__ inline v8f wmma4(v2f a, v2f b, v8f c) {
    // D(16x16,f32) = A(16x4,f32) x B(4x16,f32) + C
    return __builtin_amdgcn_wmma_f32_16x16x4_f32(
        /*neg_a=*/false, a, /*neg_b=*/false, b,
        /*c_mod=*/(short)0, c, /*reuse_a=*/false, /*reuse_b=*/false);
}

__global__ __launch_bounds__(256)
void filtered_lrelu_fused(const float* __restrict__ x,
                          const float* __restrict__ bias,
                          const float* __restrict__ fup,
                          const float* __restrict__ fdn,
                          float* __restrict__ out,
                          int nch)
{
    __shared__ float smem[SMEM_FLOATS];
    float* XT3 = smem;            // X (pitch P1) then T3 (pitch P2)
    float* T1  = smem + OFF_T1;   // 144 x P1
    float* T2  = smem + OFF_T2;   // 144 x P2
    float* FG  = smem + OFF_FG;   // f[12], g[12]

    const int tid  = threadIdx.x;
    const int lane = tid & 31;
    const int wave = tid >> 5;
    const int half = lane >> 4;   // 0: lanes 0-15, 1: lanes 16-31
    const int lr   = lane & 15;

    const int slice = blockIdx.x;
    const int ch    = slice % nch;
    const float bv  = bias[ch];
    const float* xs = x   + (size_t)slice * 4096;
    float*       os = out + (size_t)slice * 4096;

    // ---- load filters + input tile into LDS -------------------------------
    if (tid < 12)       FG[tid]      = fup[tid];
    else if (tid < 24)  FG[tid]      = fdn[tid - 12];

    // 64x64 fp32 input: 1024 float4 loads across 256 threads
    {
        const float4* xv = (const float4*)xs;
        #pragma unroll
        for (int k = 0; k < 4; ++k) {
            int i4  = tid + k * 256;        // float4 index
            float4 v = xv[i4];
            int row = i4 >> 4;
            int col = (i4 & 15) << 2;
            float* d = XT3 + row * P1 + col;
            d[0] = v.x; d[1] = v.y; d[2] = v.z; d[3] = v.w;
        }
    }
    __syncthreads();

    // ---- per-lane banded-filter fragment constants ------------------------
    // Up band:   U[i][k]  = 2*f[i + 10 - 2k]   (2 folded per direction -> gain 4)
    // Down band: Wd[i][j] = g[11 - j + 2i]
    // With the f32 16x16x4 lane layouts, the A-side fragment of the band and
    // the B-side fragment of its transpose are numerically identical, so one
    // table serves both the H-stage (A-side) and W-stage (B-side).
    const float* F = FG;
    const float* G = FG + 12;
    float upc[4][2];
    float dnc[11][2];
    #pragma unroll
    for (int c = 0; c < 4; ++c)
        #pragma unroll
        for (int v = 0; v < 2; ++v) {
            int k  = 4 * c + 2 * half + v;
            int ix = lr + 10 - 2 * k;
            upc[c][v] = (ix >= 0 && ix <= 11) ? 2.0f * F[ix] : 0.0f;
        }
    #pragma unroll
    for (int c = 0; c < 11; ++c)
        #pragma unroll
        for (int v = 0; v < 2; ++v) {
            int j  = 4 * c + 2 * half + v;
            int ix = 11 - j + 2 * lr;
            dnc[c][v] = (ix >= 0 && ix <= 11) ? G[ix] : 0.0f;
        }

    // ---- Stage A: up-H.  T1[o][c] = sum_s 2f[o-2s] X[s][c].  9x4 tiles ----
    for (int t = wave; t < 36; t += WAVES) {
        int ot = t >> 2, ct = t & 3;
        int S0 = 8 * ot - 5;                 // first X row for this tile
        int colB = 16 * ct + lr;
        v8f acc = {};
        #pragma unroll
        for (int c = 0; c < 4; ++c) {
            v2f a; a[0] = upc[c][0]; a[1] = upc[c][1];
            v2f bb;
            #pragma unroll
            for (int v = 0; v < 2; ++v) {
                int r  = S0 + 4 * c + 2 * half + v;
                int rc = r < 0 ? 0 : (r > 63 ? 63 : r);
                float val = XT3[rc * P1 + colB];
                bb[v] = (r >= 0 && r < 64) ? val : 0.0f;
            }
            acc = wmma4(a, bb, acc);
        }
        #pragma unroll
        for (int v = 0; v < 8; ++v)
            T1[(16 * ot + v + 8 * half) * P1 + colB] = acc[v];
    }
    __syncthreads();

    // ---- Stage B: up-W + bias/lrelu/clamp.  T2 = T1 x U^T.  9x9 tiles -----
    for (int t = wave; t < 81; t += WAVES) {
        int rt = t / 9, wt = t - 9 * rt;
        int S0 = 8 * wt - 5;                 // first T1 col for this tile
        int rowA = 16 * rt + lr;
        v8f acc = {};
        #pragma unroll
        for (int c = 0; c < 4; ++c) {
            v2f a;
            #pragma unroll
            for (int v = 0; v < 2; ++v) {
                int cidx = S0 + 4 * c + 2 * half + v;
                int cc   = cidx < 0 ? 0 : (cidx > 63 ? 63 : cidx);
                float val = T1[rowA * P1 + cc];
                a[v] = (cidx >= 0 && cidx < 64) ? val : 0.0f;
            }
            v2f bb; bb[0] = upc[c][0]; bb[1] = upc[c][1];
            acc = wmma4(a, bb, acc);
        }
        #pragma unroll
        for (int v = 0; v < 8; ++v) {
            float y = acc[v] + bv;
            y = (y < 0.0f ? y * 0.2f : y) * 1.41421356237309515f;
            y = fminf(fmaxf(y, -256.0f), 256.0f);
            T2[(16 * rt + v + 8 * half) * P2 + 16 * wt + lr] = y;
        }
    }
    __syncthreads();

    // ---- Stage C: down-H. T3[o][c] = sum_j g[11-j+2i] T2[2o+j][c]. 4x9 ----
    for (int t = wave; t < 36; t += WAVES) {
        int ot = t / 9, ct = t - 9 * ot;
        int colB = 16 * ct + lr;
        v8f acc = {};
        #pragma unroll
        for (int c = 0; c < 11; ++c) {
            v2f a; a[0] = dnc[c][0]; a[1] = dnc[c][1];
            v2f bb;
            #pragma unroll
            for (int v = 0; v < 2; ++v) {
                int r = 32 * ot + 4 * c + 2 * half + v;   // 2O + j, <= 139 < 144
                bb[v] = T2[r * P2 + colB];
            }
            acc = wmma4(a, bb, acc);
        }
        #pragma unroll
        for (int v = 0; v < 8; ++v)
            XT3[(16 * ot + v + 8 * half) * P2 + colB] = acc[v];  // T3 reuses X region
    }
    __syncthreads();

    // ---- Stage D: down-W -> global out (64x64).  4x4 tiles ---------------
    for (int t = wave; t < 16; t += WAVES) {
        int rt = t >> 2, wt = t & 3;
        int rowA = 16 * rt + lr;
        v8f acc = {};
        #pragma unroll
        for (int c = 0; c < 11; ++c) {
            v2f a;
            #pragma unroll
            for (int v = 0; v < 2; ++v) {
                int cidx = 32 * wt + 4 * c + 2 * half + v;  // <= 139 < 144
                a[v] = XT3[rowA * P2 + cidx];
            }
            v2f bb; bb[0] = dnc[c][0]; bb[1] = dnc[c][1];
            acc = wmma4(a, bb, acc);
        }
        #pragma unroll
        for (int v = 0; v < 8; ++v)
            os[(16 * rt + v + 8 * half) * 64 + 16 * wt + lr] = acc[v];
    }
}

extern "C" void kernel_launch(void* const* d_in, const int* in_sizes, int n_in,
                              void* d_out, int out_size, void* d_ws, size_t ws_size,
                              hipStream_t stream) {
    const float* x    = (const float*)d_in[0];   // [8,512,64,64] f32
    const float* bias = (const float*)d_in[1];   // [512] f32
    const float* fup  = (const float*)d_in[2];   // [12] f32
    const float* fdn  = (const float*)d_in[3];   // [12] f32
    float* out = (float*)d_out;                  // [8,512,64,64] f32

    const int nch    = in_sizes[1];              // 512
    const int slices = in_sizes[0] / 4096;       // N*C = 4096 (64*64 per slice)

    filtered_lrelu_fused<<<dim3(slices), dim3(256), 0, stream>>>(
        x, bias, fup, fdn, out, nch);
}